// BTier_35390530519888
// MI455X (gfx1250) — compile-verified
//
#include <hip/hip_runtime.h>
#include <cmath>

typedef __attribute__((ext_vector_type(2))) float v2f;
typedef __attribute__((ext_vector_type(8))) float v8f;

#define Bq  64
#define Sq  2048
#define Iq  128
#define Hq  256
#define BDq 128
#define LDSROW 260   // 256 + 4 pad: A-frag reads land on distinct bank groups

// ---------------------------------------------------------------------------
// Kernel 1: P[row, h] = concat(nh,nr,nt)[row] @ W_x + b_h,  row = t*64 + b
// One wave = one 16-row M-tile x one 128-col half of H (8 N-tiles, v8f accs).
// ---------------------------------------------------------------------------
__global__ __launch_bounds__(256) void k_xproj(const float* __restrict__ nh,
                                               const float* __restrict__ nr,
                                               const float* __restrict__ nt,
                                               const float* __restrict__ Wx,
                                               const float* __restrict__ bh,
                                               float* __restrict__ P) {
  const int lane = threadIdx.x & 31, wv = threadIdx.x >> 5;
  const int gw    = blockIdx.x * 8 + wv;
  const int mtile = gw >> 1, ng = gw & 1;
  const int half  = lane >> 4, l16 = lane & 15;
  const int grow  = mtile * 16 + l16;         // A-matrix row for this lane
  const int t = grow >> 6, b = grow & 63;
  const size_t xbase = ((size_t)b * Sq + t) * Iq;

  v8f acc[8];
  #pragma unroll
  for (int n = 0; n < 8; ++n) {
    const float bias = bh[ng * 128 + n * 16 + l16];
    #pragma unroll
    for (int j = 0; j < 8; ++j) acc[n][j] = bias;
  }

  for (int kk = 0; kk < 96; ++kk) {           // K = 384, 4 per WMMA
    const int kb = kk * 4 + 2 * half;         // both kb,kb+1 stay in one segment
    const float* xs = (kb < Iq) ? nh : ((kb < 2 * Iq) ? nr : nt);
    const int off = kb & (Iq - 1);
    v2f a; a.x = xs[xbase + off]; a.y = xs[xbase + off + 1];
    #pragma unroll
    for (int n = 0; n < 8; ++n) {
      const int col = ng * 128 + n * 16 + l16;
      v2f bm; bm.x = Wx[(size_t)kb * Hq + col];
              bm.y = Wx[(size_t)(kb + 1) * Hq + col];
      acc[n] = __builtin_amdgcn_wmma_f32_16x16x4_f32(
          false, a, false, bm, (short)0, acc[n], false, false);
    }
  }

  #pragma unroll
  for (int n = 0; n < 8; ++n) {
    const int col = ng * 128 + n * 16 + l16;
    #pragma unroll
    for (int j = 0; j < 8; ++j) {
      const int r = mtile * 16 + j + 8 * half;
      P[(size_t)r * Hq + col] = acc[n][j];
    }
  }
}

// ---------------------------------------------------------------------------
// Kernel 2: sequential scan. Single 1024-thread workgroup, persistent over t.
// h in LDS (padded); W_h streamed from L1/L2; P[t] read (Xproj) then
// overwritten in place with h_new (becomes H_all[t]).
// Wave w: N-tile n = w&15, M-tiles {w>>4, (w>>4)+2}.
// ---------------------------------------------------------------------------
__global__ __launch_bounds__(1024) void k_scan(const float* __restrict__ h0,
                                               const float* __restrict__ Wh,
                                               float* __restrict__ P,
                                               float* __restrict__ hlast) {
  __shared__ float hsh[64 * LDSROW];
  const int tid = threadIdx.x;
  for (int i = tid; i < 64 * Hq; i += 1024)
    hsh[(i >> 8) * LDSROW + (i & 255)] = h0[i];
  __syncthreads();

  const int lane = tid & 31, wv = tid >> 5;
  const int half = lane >> 4, l16 = lane & 15;
  const int n  = wv & 15;
  const int m0 = wv >> 4;          // 0..1
  const int m1 = m0 + 2;           // 2..3
  const int col = n * 16 + l16;

  for (int t = 0; t < Sq; ++t) {
    const size_t pbase = (size_t)t * (64 * Hq);
    v8f acc0, acc1;
    #pragma unroll
    for (int j = 0; j < 8; ++j) {
      acc0[j] = P[pbase + (size_t)(m0 * 16 + j + 8 * half) * Hq + col];
      acc1[j] = P[pbase + (size_t)(m1 * 16 + j + 8 * half) * Hq + col];
    }
    for (int kk = 0; kk < 64; ++kk) {        // K = 256
      const int kb = kk * 4 + 2 * half;
      v2f a0, a1, bm;
      a0.x = hsh[(m0 * 16 + l16) * LDSROW + kb];
      a0.y = hsh[(m0 * 16 + l16) * LDSROW + kb + 1];
      a1.x = hsh[(m1 * 16 + l16) * LDSROW + kb];
      a1.y = hsh[(m1 * 16 + l16) * LDSROW + kb + 1];
      bm.x = Wh[(size_t)kb * Hq + col];
      bm.y = Wh[(size_t)(kb + 1) * Hq + col];
      acc0 = __builtin_amdgcn_wmma_f32_16x16x4_f32(
          false, a0, false, bm, (short)0, acc0, false, false);
      acc1 = __builtin_amdgcn_wmma_f32_16x16x4_f32(
          false, a1, false, bm, (short)0, acc1, false, false);
    }
    __syncthreads();                          // all reads of hsh complete
    #pragma unroll
    for (int j = 0; j < 8; ++j) {
      const int r0 = m0 * 16 + j + 8 * half;
      const int r1 = m1 * 16 + j + 8 * half;
      const float v0 = tanhf(acc0[j]);
      const float v1 = tanhf(acc1[j]);
      hsh[r0 * LDSROW + col] = v0;
      hsh[r1 * LDSROW + col] = v1;
      P[pbase + (size_t)r0 * Hq + col] = v0;  // H_all[t] for kernel 3
      P[pbase + (size_t)r1 * Hq + col] = v1;
      if (t == Sq - 1) {
        hlast[r0 * Hq + col] = v0;
        hlast[r1 * Hq + col] = v1;
      }
    }
    __syncthreads();                          // h_new visible for next step
  }
}

// ---------------------------------------------------------------------------
// Kernel 3: outs[b, t, :] = tanh(H_all[t,b,:] @ W_b + b_b)
// One wave = one 16-row M-tile x all 8 N-tiles of BD=128.
// ---------------------------------------------------------------------------
__global__ __launch_bounds__(256) void k_out(const float* __restrict__ P,
                                             const float* __restrict__ Wb,
                                             const float* __restrict__ bb,
                                             float* __restrict__ outs) {
  const int lane = threadIdx.x & 31, wv = threadIdx.x >> 5;
  const int mtile = blockIdx.x * 8 + wv;
  const int half = lane >> 4, l16 = lane & 15;
  const int arow = mtile * 16 + l16;

  v8f acc[8];
  #pragma unroll
  for (int n = 0; n < 8; ++n) {
    const float bias = bb[n * 16 + l16];
    #pragma unroll
    for (int j = 0; j < 8; ++j) acc[n][j] = bias;
  }
  for (int kk = 0; kk < 64; ++kk) {           // K = 256
    const int kb = kk * 4 + 2 * half;
    v2f a; a.x = P[(size_t)arow * Hq + kb];
           a.y = P[(size_t)arow * Hq + kb + 1];
    #pragma unroll
    for (int n = 0; n < 8; ++n) {
      const int col = n * 16 + l16;
      v2f bm; bm.x = Wb[(size_t)kb * BDq + col];
              bm.y = Wb[(size_t)(kb + 1) * BDq + col];
      acc[n] = __builtin_amdgcn_wmma_f32_16x16x4_f32(
          false, a, false, bm, (short)0, acc[n], false, false);
    }
  }
  #pragma unroll
  for (int n = 0; n < 8; ++n) {
    const int col = n * 16 + l16;
    #pragma unroll
    for (int j = 0; j < 8; ++j) {
      const int row = mtile * 16 + j + 8 * half;   // = t*64 + b
      const int tt = row >> 6, bb_ = row & 63;
      outs[(size_t)bb_ * Sq * BDq + (size_t)tt * BDq + col] = tanhf(acc[n][j]);
    }
  }
}

// ---------------------------------------------------------------------------
extern "C" void kernel_launch(void* const* d_in, const int* in_sizes, int n_in,
                              void* d_out, int out_size, void* d_ws, size_t ws_size,
                              hipStream_t stream) {
  const float* nh = (const float*)d_in[0];
  const float* nr = (const float*)d_in[1];
  const float* nt = (const float*)d_in[2];
  const float* h0 = (const float*)d_in[3];
  const float* Wx = (const float*)d_in[4];
  const float* Wh = (const float*)d_in[5];
  const float* bh = (const float*)d_in[6];
  const float* Wb = (const float*)d_in[7];
  const float* bb = (const float*)d_in[8];

  float* P     = (float*)d_ws;                         // S*B*H fp32 = 128 MB
  float* outs  = (float*)d_out;                        // [B,S,BD]
  float* hlast = outs + (size_t)Bq * Sq * BDq;         // [B,H]

  // Phase 1: 8192 M-tiles x 2 col-halves = 16384 waves (8 waves/block)
  k_xproj<<<2048, 256, 0, stream>>>(nh, nr, nt, Wx, bh, P);
  // Phase 2: one persistent workgroup, 32 waves, 2048 sequential steps
  k_scan<<<1, 1024, 0, stream>>>(h0, Wh, P, hlast);
  // Phase 3: 8192 M-tiles, 8 waves/block
  k_out<<<1024, 256, 0, stream>>>(P, Wb, bb, outs);
}